// GCN_47227460387599
// MI455X (gfx1250) — compile-verified
//
#include <hip/hip_runtime.h>

typedef __attribute__((ext_vector_type(2))) float v2f;
typedef __attribute__((ext_vector_type(8))) float v8f;

#define K_DIM 128  // every layer's input width is 128

// ---------------- degree / normalization precompute ----------------

__global__ void fill_ones_k(float* __restrict__ p, int n) {
  int i = blockIdx.x * blockDim.x + threadIdx.x;
  if (i < n) p[i] = 1.0f;  // self-loop contributes 1 to every degree
}

__global__ void count_deg_k(const int* __restrict__ dst, float* __restrict__ deg, int e) {
  int i = blockIdx.x * blockDim.x + threadIdx.x;
  if (i < e) atomicAdd(&deg[dst[i]], 1.0f);
}

__global__ void rsqrt_ip_k(float* __restrict__ p, int n) {
  int i = blockIdx.x * blockDim.x + threadIdx.x;
  if (i < n) p[i] = rsqrtf(p[i]);
}

// ---------------- f32 WMMA GEMM: T[n x (NT*16)] = A[n x 128] @ W[128 x (NT*16)] ----
// One wave computes a 16-row strip across all NT column tiles.
// V_WMMA_F32_16X16X4_F32 operand layout (wave32):
//   A 16x4 : lanes 0-15 hold M=lane, {v0,v1}={K0,K1}; lanes 16-31 hold {K2,K3}
//   B 4x16 : lanes 0-15 hold N=lane, {v0,v1}={K0,K1}; lanes 16-31 hold {K2,K3}
//   C 16x16: lanes 0-15 v[r]=C[M=r][N=lane]; lanes 16-31 v[r]=C[M=8+r][N=lane-16]

template<int NT>
__global__ __launch_bounds__(128) void gemm_wmma_f32(
    const float* __restrict__ A, const float* __restrict__ W,
    float* __restrict__ T, int nrows)
{
  constexpr int MOUT = NT * 16;
  const int lane  = threadIdx.x & 31;
  const int tile  = blockIdx.x * (blockDim.x >> 5) + (threadIdx.x >> 5);
  if (tile * 16 >= nrows) return;            // wave-uniform exit, EXEC stays all-1 in body
  const int khalf = lane >> 4;               // 0: K0/K1, 1: K2/K3
  const int lcol  = lane & 15;

  const float* arow = A + (size_t)(tile * 16 + lcol) * K_DIM + 2 * khalf;

  v8f acc[NT] = {};

  for (int k0 = 0; k0 < K_DIM; k0 += 4) {
    v2f a;
    a.x = arow[k0];
    a.y = arow[k0 + 1];
    const float* b0 = W + (size_t)(k0 + 2 * khalf) * MOUT + lcol;
#pragma unroll
    for (int t = 0; t < NT; ++t) {
      v2f b;
      b.x = b0[t * 16];
      b.y = b0[t * 16 + MOUT];
      acc[t] = __builtin_amdgcn_wmma_f32_16x16x4_f32(
          /*neg_a=*/false, a, /*neg_b=*/false, b,
          /*c_mod=*/(short)0, acc[t], /*reuse_a=*/false, /*reuse_b=*/false);
    }
  }

  float* crow = T + (size_t)(tile * 16 + khalf * 8) * MOUT + lcol;
#pragma unroll
  for (int t = 0; t < NT; ++t) {
#pragma unroll
    for (int r = 0; r < 8; ++r)
      crow[(size_t)r * MOUT + t * 16] = acc[t][r];
  }
}

// ---------------- aggregation ----------------

// acc[i][f] = T[i][f] * dinv[i]^2 + bias[f]   (self-loop term + bias)
template<int F>
__global__ void self_bias_k(const float* __restrict__ T, const float* __restrict__ dinv,
                            const float* __restrict__ bias, float* __restrict__ acc, int n)
{
  size_t i = (size_t)blockIdx.x * blockDim.x + threadIdx.x;
  if (i >= (size_t)n * F) return;
  int node = (int)(i / F);
  int f    = (int)(i % F);
  float di = dinv[node];
  acc[i] = T[i] * di * di + bias[f];
}

// one wave per edge: acc[dst] += T[src] * dinv[src]*dinv[dst]
template<int F>
__global__ void scatter_edges_k(const float* __restrict__ T, const int* __restrict__ src,
                                const int* __restrict__ dst, const float* __restrict__ dinv,
                                float* __restrict__ acc, int e)
{
  int gid  = blockIdx.x * blockDim.x + threadIdx.x;
  int edge = gid >> 5;
  int lane = gid & 31;
  if (edge >= e) return;
  int s = src[edge], d = dst[edge];
  float w = dinv[s] * dinv[d];
  constexpr int V = F / 32;                  // 4 floats/lane (F=128) or 2 (F=64)
  const float* tp = T   + (size_t)s * F + lane * V;
  float*       op = acc + (size_t)d * F + lane * V;
  float vals[V];
#pragma unroll
  for (int i = 0; i < V; ++i) vals[i] = tp[i];
#pragma unroll
  for (int i = 0; i < V; ++i) atomicAdd(&op[i], vals[i] * w);
}

__global__ void relu_ip_k(float* __restrict__ p, size_t n) {
  size_t i = (size_t)blockIdx.x * blockDim.x + threadIdx.x;
  if (i < n) p[i] = fmaxf(p[i], 0.0f);
}

// ---------------- launcher ----------------

extern "C" void kernel_launch(void* const* d_in, const int* in_sizes, int n_in,
                              void* d_out, int out_size, void* d_ws, size_t ws_size,
                              hipStream_t stream)
{
  const float* x     = (const float*)d_in[0];
  const int*   ei    = (const int*)  d_in[1];
  const float* W_in  = (const float*)d_in[2];
  const float* b_in  = (const float*)d_in[3];
  const float* W_h0  = (const float*)d_in[4];
  const float* b_h0  = (const float*)d_in[5];
  const float* W_h1  = (const float*)d_in[6];
  const float* b_h1  = (const float*)d_in[7];
  const float* W_out = (const float*)d_in[8];
  const float* b_out = (const float*)d_in[9];

  const int FIN = 128, HID = 128, COUT = 64;
  const int N = in_sizes[0] / FIN;
  const int E = in_sizes[1] / 2;
  const int* src = ei;
  const int* dst = ei + E;

  float* ws   = (float*)d_ws;
  float* dinv = ws;                          // N
  float* T    = ws + N;                      // N*128 (GEMM output scratch)
  float* H1   = T  + (size_t)N * HID;        // N*128
  float* H2   = H1 + (size_t)N * HID;        // N*128
  float* out  = (float*)d_out;               // N*64

  // --- degree / dinv ---
  fill_ones_k<<<(N + 255) / 256, 256, 0, stream>>>(dinv, N);
  count_deg_k<<<(E + 255) / 256, 256, 0, stream>>>(dst, dinv, E);
  rsqrt_ip_k <<<(N + 255) / 256, 256, 0, stream>>>(dinv, N);

  const int tiles   = (N + 15) / 16;
  const int gblocks = (tiles + 3) / 4;       // 4 waves / block
  const int eblocks = (E * 32 + 255) / 256;  // 1 wave / edge
  const size_t nh = (size_t)N * HID;
  const size_t nc = (size_t)N * COUT;
  const int nhb = (int)((nh + 255) / 256);
  const int ncb = (int)((nc + 255) / 256);

  // layer 1: x -> H1
  gemm_wmma_f32<8><<<gblocks, 128, 0, stream>>>(x, W_in, T, N);
  self_bias_k<128><<<nhb, 256, 0, stream>>>(T, dinv, b_in, H1, N);
  scatter_edges_k<128><<<eblocks, 256, 0, stream>>>(T, src, dst, dinv, H1, E);
  relu_ip_k<<<nhb, 256, 0, stream>>>(H1, nh);

  // layer 2: H1 -> H2
  gemm_wmma_f32<8><<<gblocks, 128, 0, stream>>>(H1, W_h0, T, N);
  self_bias_k<128><<<nhb, 256, 0, stream>>>(T, dinv, b_h0, H2, N);
  scatter_edges_k<128><<<eblocks, 256, 0, stream>>>(T, src, dst, dinv, H2, E);
  relu_ip_k<<<nhb, 256, 0, stream>>>(H2, nh);

  // layer 3: H2 -> H1
  gemm_wmma_f32<8><<<gblocks, 128, 0, stream>>>(H2, W_h1, T, N);
  self_bias_k<128><<<nhb, 256, 0, stream>>>(T, dinv, b_h1, H1, N);
  scatter_edges_k<128><<<eblocks, 256, 0, stream>>>(T, src, dst, dinv, H1, E);
  relu_ip_k<<<nhb, 256, 0, stream>>>(H1, nh);

  // layer 4: H1 -> out (64 wide)
  gemm_wmma_f32<4><<<gblocks, 128, 0, stream>>>(H1, W_out, T, N);
  self_bias_k<64><<<ncb, 256, 0, stream>>>(T, dinv, b_out, out, N);
  scatter_edges_k<64><<<eblocks, 256, 0, stream>>>(T, src, dst, dinv, out, E);
  relu_ip_k<<<ncb, 256, 0, stream>>>(out, nc);
}